// RegionLoss_47098611368203
// MI455X (gfx1250) — compile-verified
//
#include <hip/hip_runtime.h>
#include <cstdint>

// ---------------------------------------------------------------------------
// RegionLoss for MI455X (gfx1250, wave32).
// Roofline: only dense pass is the fm-weighted BCE over the conf channel:
// 32*5*96*96 floats = 5.9 MB  ->  ~0.25us at 23.3 TB/s. Everything else is a
// <=1600-cell scatter/gather. So: minimize passes, stream the conf planes via
// CDNA5 async global->LDS (ASYNCcnt), do masked terms on a compact cell list,
// and use V_WMMA_F32_16X16X4_F32 (f32, lossless) for the final partial-sum
// reduction.
// ---------------------------------------------------------------------------

#define B_ 32
#define A_ 5
#define C_ 8
#define H_ 96
#define W_ 96
#define O_ 50
#define NCH (7 + C_)                       // 15 channels per anchor
#define PLANE (H_ * W_)                    // 9216
#define NPLANES (B_ * A_)                  // 160
#define WORDS_PER_PLANE (PLANE / 32)       // 288
#define WORDS_PER_BATCH (A_ * WORDS_PER_PLANE)  // 1440
#define MAXE 64
#define IGNORE_THR 0.6f
#define CONF_THR 0.5f

// Workspace overlay (~440 KB). Every field read in a call is written earlier
// in the same call, so the harness's 0xAA poison is never observed.
struct Ws {
  uint32_t confW[B_ * WORDS_PER_BATCH];    // conf bitmap (1 = conf true)
  uint32_t maskW[B_ * WORDS_PER_BATCH];    // mask bitmap
  int      ecnt[B_];                       // masked-cell count per batch
  int      ecell[B_ * MAXE];               // cell index a*PLANE + ax*W + ay
  int      ecls[B_ * MAXE];                // class id per masked cell
  float    evals[B_ * MAXE * 6];           // tx,ty,tl,tw,tim,tre
  int      ngt[B_];
  int      ncor[B_];
  float    bceP[NPLANES];                  // per-plane fm-BCE partial
  float    nfP[NPLANES];                   // per-plane fm count
  float    csums[16];                      // kernel-C sums + nt
};

__device__ __forceinline__ float sigmoidf_(float x) { return 1.0f / (1.0f + expf(-x)); }

__device__ __forceinline__ float box_iou(float gx, float gy, float gl, float gw,
                                         float bx, float by, float bl, float bw) {
  float min_x = fminf(gx - gl * 0.5f, bx - bl * 0.5f);
  float max_x = fmaxf(gx + gl * 0.5f, bx + bl * 0.5f);
  float min_y = fminf(gy - gw * 0.5f, by - bw * 0.5f);
  float max_y = fmaxf(gy + gw * 0.5f, by + bw * 0.5f);
  float uw = max_y - min_y;
  float uh = max_x - min_x;
  float iw = gw + bw - uw;
  float il = gl + bl - uh;
  float inter = (iw <= 0.0f || il <= 0.0f) ? 0.0f : iw * il;
  float uni = gw * gl + bw * bl - inter;
  return inter / uni;
}

// ---------------------------------------------------------------------------
// Kernel A: per-batch sequential scatter (exactly reproduces scan order).
// Builds conf/mask bitmaps in LDS, compact masked-cell list in ws, and
// n_gt/n_cor from fmap_preds gathered at the scattered cells.
// ---------------------------------------------------------------------------
__global__ void __launch_bounds__(64) kA(const float* __restrict__ out,
                                         const float* __restrict__ tgt,
                                         const float* __restrict__ anch,
                                         Ws* __restrict__ ws) {
  __shared__ uint32_t confW[WORDS_PER_BATCH];
  __shared__ uint32_t maskW[WORDS_PER_BATCH];
  const int b = blockIdx.x;
  for (int i = threadIdx.x; i < WORDS_PER_BATCH; i += blockDim.x) {
    confW[i] = 0xFFFFFFFFu;   // conf starts all-true
    maskW[i] = 0u;
  }
  __syncthreads();

  if (threadIdx.x == 0) {
    int cnt = 0, ngt = 0, ncor = 0;
    for (int o = 0; o < O_; ++o) {
      const float* t = tgt + ((size_t)b * O_ + o) * 7;
      float cls = t[0];
      if (cls == 0.0f) break;                 // cumprod validity
      float gx = t[1] * (float)H_, gy = t[2] * (float)W_;
      float gl = t[3] * (float)H_, gw = t[4] * (float)W_;

      // shape-only IoU vs anchors (boxes centered at origin):
      // union_h = max(gl, al), union_w = max(gw, aw)
      float ious[A_];
      int best = 0; float bestv = -1.0f;
      for (int a = 0; a < A_; ++a) {
        float al = anch[a * 2 + 0], aw = anch[a * 2 + 1];
        float iw = gw + aw - fmaxf(gw, aw);
        float il = gl + al - fmaxf(gl, al);
        float inter = (iw <= 0.0f || il <= 0.0f) ? 0.0f : iw * il;
        float uni = gw * gl + aw * al - inter;
        float v = inter / uni;
        ious[a] = v;
        if (v > bestv) { bestv = v; best = a; }   // first max, like argmax
      }
      int ax = (int)gx; ax = ax < 0 ? 0 : (ax > H_ - 1 ? H_ - 1 : ax);
      int ay = (int)gy; ay = ay < 0 ? 0 : (ay > W_ - 1 ? W_ - 1 : ay);

      // conf clear at all anchors with iou > thr, then set at best anchor
      for (int a = 0; a < A_; ++a) {
        if (ious[a] > IGNORE_THR) {
          int bi = a * PLANE + ax * W_ + ay;
          confW[bi >> 5] &= ~(1u << (bi & 31));
        }
      }
      int bi = best * PLANE + ax * W_ + ay;
      confW[bi >> 5] |= 1u << (bi & 31);
      maskW[bi >> 5] |= 1u << (bi & 31);

      // compact entry list (last writer wins, preserves scan semantics)
      int slot = -1;
      for (int e = 0; e < cnt; ++e)
        if (ws->ecell[b * MAXE + e] == bi) { slot = e; break; }
      if (slot < 0) { slot = cnt++; ws->ecell[b * MAXE + slot] = bi; }
      float* ev = &ws->evals[((size_t)b * MAXE + slot) * 6];
      ev[0] = gx - (float)ax;
      ev[1] = gy - (float)ay;
      ev[2] = logf(gl / anch[best * 2 + 0]);
      ev[3] = logf(gw / anch[best * 2 + 1]);
      ev[4] = t[5];
      ev[5] = t[6];
      int cid = (int)cls; cid = cid < 0 ? 0 : (cid > C_ - 1 ? C_ - 1 : cid);
      ws->ecls[b * MAXE + slot] = cid;

      // fmap_preds at (b, best, ax, ay) -> piou / correct
      const float* op = out + ((size_t)b * A_ * NCH + (size_t)best * NCH) * PLANE
                        + ax * W_ + ay;
      float px = sigmoidf_(op[0 * PLANE]) + (float)ax;
      float py = sigmoidf_(op[1 * PLANE]) + (float)ay;
      float pl = expf(op[2 * PLANE]) * anch[best * 2 + 0];
      float pw = expf(op[3 * PLANE]) * anch[best * 2 + 1];
      float pc = sigmoidf_(op[6 * PLANE]);
      float piou = box_iou(gx, gy, gl, gw, px, py, pl, pw);
      ngt += 1;
      if (piou > 0.5f && pc > CONF_THR) ncor += 1;
    }
    ws->ecnt[b] = cnt;
    ws->ngt[b] = ngt;
    ws->ncor[b] = ncor;
  }
  __syncthreads();
  for (int i = threadIdx.x; i < WORDS_PER_BATCH; i += blockDim.x) {
    ws->confW[(size_t)b * WORDS_PER_BATCH + i] = confW[i];
    ws->maskW[(size_t)b * WORDS_PER_BATCH + i] = maskW[i];
  }
}

// ---------------------------------------------------------------------------
// Kernel B: one block per (b,a) conf plane (contiguous 36 KB). Streams the
// plane through LDS with CDNA5 async global->LDS (ASYNCcnt path), computes
// the fm-weighted BCE and the fm count, deterministic LDS tree reduce.
// ---------------------------------------------------------------------------
#define KB_THREADS 256
#define KB_CHUNK 1024   // floats per round = 256 lanes * 4 (b128 each)

__global__ void __launch_bounds__(KB_THREADS) kB(const float* __restrict__ out,
                                                 Ws* __restrict__ ws) {
  __shared__ __align__(16) float buf[KB_CHUNK];
  __shared__ float rs[KB_THREADS];
  __shared__ float rc[KB_THREADS];

  const int p = blockIdx.x;                 // plane = b*A_ + a
  const int b = p / A_, a = p % A_;
  const float* src = out + ((size_t)b * A_ * NCH + (size_t)a * NCH + 6) * PLANE;
  const uint32_t* cw = &ws->confW[(size_t)p * WORDS_PER_PLANE];
  const uint32_t* mw = &ws->maskW[(size_t)p * WORDS_PER_PLANE];
  const int tid = threadIdx.x;

  float s = 0.0f, cnt = 0.0f;
  // group-segment (LDS) offset for this lane's 16B slot
  uint32_t ldsOff = (uint32_t)(uintptr_t)(&buf[tid * 4]);

  const int rounds = PLANE / KB_CHUNK;      // 9
  for (int c = 0; c < rounds; ++c) {
    const float* g = src + c * KB_CHUNK + tid * 4;
    if (c + 1 < rounds)
      __builtin_prefetch(g + KB_CHUNK, 0, 1);       // global_prefetch_b8
    // CDNA5 async copy: 16 B/lane global -> LDS, tracked by ASYNCcnt
    asm volatile("global_load_async_to_lds_b128 %0, %1, off"
                 :: "v"(ldsOff), "v"((uint64_t)(uintptr_t)g) : "memory");
    asm volatile("s_wait_asynccnt 0" ::: "memory");
    __syncthreads();
    #pragma unroll
    for (int j = 0; j < 4; ++j) {
      int idx = c * KB_CHUNK + tid * 4 + j;         // cell index in plane
      float x = buf[tid * 4 + j];
      uint32_t w = (uint32_t)idx >> 5, bit = (uint32_t)idx & 31u;
      int cbit = (cw[w] >> bit) & 1;
      int mbit = (mw[w] >> bit) & 1;
      if (cbit != mbit) {                           // fm cell
        float pr = sigmoidf_(x);
        float lp = fmaxf(logf(pr), -100.0f);
        float ln = fmaxf(logf(1.0f - pr), -100.0f);
        s += -(mbit ? lp : ln);                     // t == mask bit
        cnt += 1.0f;
      }
    }
    __syncthreads();
  }

  rs[tid] = s; rc[tid] = cnt;
  __syncthreads();
  for (int st = KB_THREADS / 2; st > 0; st >>= 1) {
    if (tid < st) { rs[tid] += rs[tid + st]; rc[tid] += rc[tid + st]; }
    __syncthreads();
  }
  if (tid == 0) { ws->bceP[p] = rs[0]; ws->nfP[p] = rc[0]; }
}

// ---------------------------------------------------------------------------
// Kernel C: masked-cell terms (<=1600 cells): mmse x/y/l/w/im/re, m-weighted
// conf BCE (t==1), cross-entropy on sigmoid(cls), and nt.
// ---------------------------------------------------------------------------
__global__ void __launch_bounds__(256) kC(const float* __restrict__ out,
                                          Ws* __restrict__ ws) {
  __shared__ float red[256];
  const int tid = threadIdx.x;
  float sx = 0, sy = 0, sl = 0, sw = 0, sim = 0, sre = 0, sconf = 0, sce = 0, snt = 0;

  for (int b = 0; b < B_; ++b) {
    int cnt = ws->ecnt[b];
    if (tid < cnt) {
      int cell = ws->ecell[b * MAXE + tid];
      int a = cell / PLANE;
      int rem = cell - a * PLANE;
      const float* op = out + ((size_t)b * A_ * NCH + (size_t)a * NCH) * PLANE + rem;
      const float* ev = &ws->evals[((size_t)b * MAXE + tid) * 6];
      float o[NCH];
      #pragma unroll
      for (int ch = 0; ch < NCH; ++ch) o[ch] = op[ch * PLANE];

      float d0 = sigmoidf_(o[0]) - ev[0];
      float d1 = sigmoidf_(o[1]) - ev[1];
      float d2 = o[2] - ev[2];
      float d3 = o[3] - ev[3];
      float d4 = o[4] - ev[4];
      float d5 = o[5] - ev[5];
      sx += d0 * d0; sy += d1 * d1; sl += d2 * d2;
      sw += d3 * d3; sim += d4 * d4; sre += d5 * d5;

      float pc = sigmoidf_(o[6]);
      sconf += -fmaxf(logf(pc), -100.0f);           // t == 1 term only

      float ocls[C_], mx = -1e30f;
      #pragma unroll
      for (int i = 0; i < C_; ++i) { ocls[i] = sigmoidf_(o[7 + i]); mx = fmaxf(mx, ocls[i]); }
      float se = 0.0f;
      #pragma unroll
      for (int i = 0; i < C_; ++i) se += expf(ocls[i] - mx);
      float lse = mx + logf(se);
      sce += lse - ocls[ws->ecls[b * MAXE + tid]];
      snt += 1.0f;
    }
  }

  float vals[9] = {sx, sy, sl, sw, sim, sre, sconf, sce, snt};
  for (int q = 0; q < 9; ++q) {
    red[tid] = vals[q];
    __syncthreads();
    for (int st = 128; st > 0; st >>= 1) {
      if (tid < st) red[tid] += red[tid + st];
      __syncthreads();
    }
    if (tid == 0) ws->csums[q] = red[0];
    __syncthreads();
  }
}

// ---------------------------------------------------------------------------
// Kernel D: combine. The 160-entry partial arrays are reduced with
// V_WMMA_F32_16X16X4_F32: partials as a 16x4 f32 A-matrix, all-ones B ->
// D[m,n] = rowsum_m (f32, lossless). Three chunks chained via the C operand,
// then 8 per-lane adds + one xor-16 shuffle finish the sum. One full wave,
// no divergence before WMMA (EXEC all ones, per ISA restriction).
// ---------------------------------------------------------------------------
typedef float v2f __attribute__((ext_vector_type(2)));
typedef float v8f __attribute__((ext_vector_type(8)));

__device__ __forceinline__ float wmma_reduce160(const float* __restrict__ part) {
  const int lane = threadIdx.x;  // 0..31, full wave
#if __has_builtin(__builtin_amdgcn_wmma_f32_16x16x4_f32)
  v8f acc = {};
  v2f ones; ones.x = 1.0f; ones.y = 1.0f;
  const int m = (lane < 16) ? lane : lane - 16;
  const int kb = (lane < 16) ? 0 : 2;
  #pragma unroll
  for (int c = 0; c < 3; ++c) {                     // 3*64 = 192 >= 160
    int i0 = c * 64 + m * 4 + kb + 0;
    int i1 = c * 64 + m * 4 + kb + 1;
    v2f a;
    a.x = (i0 < NPLANES) ? part[i0] : 0.0f;
    a.y = (i1 < NPLANES) ? part[i1] : 0.0f;
    acc = __builtin_amdgcn_wmma_f32_16x16x4_f32(
        /*neg_a=*/false, a, /*neg_b=*/false, ones,
        /*c_mod=*/(short)0, acc, /*reuse_a=*/false, /*reuse_b=*/false);
  }
  float s = 0.0f;
  #pragma unroll
  for (int v = 0; v < 8; ++v) s += acc[v];          // rows 0-7 (lanes<16) / 8-15
  s += __shfl_xor(s, 16, 32);                       // total in every lane
  return s;
#else
  float s = 0.0f;
  if (lane == 0)
    for (int i = 0; i < NPLANES; ++i) s += part[i];
  return __shfl(s, 0, 32);
#endif
}

__global__ void __launch_bounds__(32) kD(Ws* __restrict__ ws, float* __restrict__ outp) {
  float bce_fm = wmma_reduce160(ws->bceP);
  float nf     = wmma_reduce160(ws->nfP);
  if (threadIdx.x == 0) {
    float nt = ws->csums[8];
    float inv_nt = 1.0f / nt;
    float loss = (ws->csums[0] + ws->csums[1] + ws->csums[2] +
                  ws->csums[3] + ws->csums[4] + ws->csums[5]) * inv_nt;   // x,y,l,w,im,re
    loss += bce_fm / nf + ws->csums[6] * inv_nt;                          // loss_conf
    loss += (1.0f / (float)B_) * ws->csums[7] * inv_nt;                   // loss_cls
    int ngt = 0, ncor = 0;
    for (int b = 0; b < B_; ++b) { ngt += ws->ngt[b]; ncor += ws->ncor[b]; }
    outp[0] = loss;
    outp[1] = (float)ngt;
    outp[2] = (float)ncor;
  }
}

// ---------------------------------------------------------------------------
extern "C" void kernel_launch(void* const* d_in, const int* in_sizes, int n_in,
                              void* d_out, int out_size, void* d_ws, size_t ws_size,
                              hipStream_t stream) {
  const float* out  = (const float*)d_in[0];   // (32, 75, 96, 96)
  const float* tgt  = (const float*)d_in[1];   // (32, 50, 7)
  const float* anch = (const float*)d_in[2];   // (5, 2)
  Ws* ws = (Ws*)d_ws;                          // ~440 KB scratch
  float* o = (float*)d_out;                    // [loss, n_gt, n_cor]

  kA<<<B_, 64, 0, stream>>>(out, tgt, anch, ws);
  kB<<<NPLANES, KB_THREADS, 0, stream>>>(out, ws);
  kC<<<1, 256, 0, stream>>>(out, ws);
  kD<<<1, 32, 0, stream>>>(ws, o);
}